// ChamferDistanceLoss_77232101916760
// MI455X (gfx1250) — compile-verified
//
#include <hip/hip_runtime.h>

typedef float v2f __attribute__((ext_vector_type(2)));
typedef float v8f __attribute__((ext_vector_type(8)));

#define CCH 16            // column chunks per pass
#define WPB 8             // waves per block (256 threads, wave32)

// Per cloud, write the WMMA B-operand layout directly:
//   Pop[j]     = (-2x, -2y)          (K0, K1 for lanes 0-15)
//   Pop[n + j] = (-2z, ||p||^2)      (K2, K3 for lanes 16-31)
// and init row-min bit arrays to +inf.
__global__ void chamfer_prep(const float* __restrict__ p1, int n1,
                             const float* __restrict__ p2, int n2,
                             float2* __restrict__ P1op, float2* __restrict__ P2op,
                             unsigned* __restrict__ m1, unsigned* __restrict__ m2) {
    int i = blockIdx.x * blockDim.x + threadIdx.x;
    if (i < n1) {
        float x = p1[3 * i + 0], y = p1[3 * i + 1], z = p1[3 * i + 2];
        P1op[i]      = make_float2(-2.0f * x, -2.0f * y);
        P1op[n1 + i] = make_float2(-2.0f * z, x * x + y * y + z * z);
        m1[i] = 0x7F800000u;   // +inf
    }
    if (i < n2) {
        float x = p2[3 * i + 0], y = p2[3 * i + 1], z = p2[3 * i + 2];
        P2op[i]      = make_float2(-2.0f * x, -2.0f * y);
        P2op[n2 + i] = make_float2(-2.0f * z, x * x + y * y + z * z);
        m2[i] = 0x7F800000u;   // +inf
    }
}

// dist(i,j) = ||a||^2 - 2 a.b + ||b||^2 via one V_WMMA_F32_16X16X4_F32 per tile:
//   A row i = (x, y, z, 1)          (recovered from the packed operand array)
//   B col j = (-2x, -2y, -2z, ||b||^2)   (one b64 load, already in lane layout)
//   C[i][j] = ||a_i||^2             (row-constant -> loop-invariant accumulator)
// Inner loop per tile: 1 load + 1 wmma + 8 min.
__global__ __launch_bounds__(256) void chamfer_rowmin(
    const float2* __restrict__ Aop, int nA,
    const float2* __restrict__ Bop, int nB,
    unsigned* __restrict__ rowminBits) {
    const int wave = threadIdx.x >> 5;
    const int lane = threadIdx.x & 31;
    const int l    = lane & 15;         // M (for A) / N (for B,C) index in tile
    const bool hi  = lane >= 16;        // upper half -> K=2,3 (A/B), M+8 (C/D)

    const int unit     = blockIdx.x * WPB + wave;
    const int rowBlock = unit / CCH;
    const int chunk    = unit % CCH;
    const int r0       = rowBlock * 16;
    const int colsPerChunk = nB / CCH;  // multiple of 16
    const int c0       = chunk * colsPerChunk;
    const int tiles    = colsPerChunk >> 4;

    // A operand (fixed): lanes 0-15: (x, y); lanes 16-31: (z, 1).
    // Recover coords from the packed (-2*) arrays (one-time cost).
    float2 a0 = Aop[r0 + l];
    float2 a1 = Aop[nA + r0 + l];
    v2f a;
    a.x = hi ? (-0.5f * a1.x) : (-0.5f * a0.x);
    a.y = hi ? 1.0f : (-0.5f * a0.y);

    // C accumulator (fixed): c[r] = ||a_{r0+r+8*hi}||^2 (lane-half uniform).
    v8f c;
    const int cbase = nA + r0 + (hi ? 8 : 0);
#pragma unroll
    for (int r = 0; r < 8; ++r) c[r] = Aop[cbase + r].y;

    v8f vmin;
#pragma unroll
    for (int r = 0; r < 8; ++r) vmin[r] = __builtin_inff();

    // B operand stream: per lane one b64 per tile, already in WMMA layout.
    const float2* bptr = Bop + (hi ? nB : 0) + c0 + l;
    for (int t = 0; t < tiles; ++t) {
        float2 bv = bptr[t * 16];
        v2f b;
        b.x = bv.x;
        b.y = bv.y;
        v8f d = __builtin_amdgcn_wmma_f32_16x16x4_f32(
            /*neg_a=*/false, a, /*neg_b=*/false, b,
            /*c_mod=*/(short)0, c, /*reuse_a=*/false, /*reuse_b=*/false);
#pragma unroll
        for (int r = 0; r < 8; ++r) vmin[r] = fminf(vmin[r], d[r]);
    }

    // Min-reduce across each 16-lane half (columns of the running tile mins).
#pragma unroll
    for (int m = 8; m >= 1; m >>= 1) {
#pragma unroll
        for (int r = 0; r < 8; ++r)
            vmin[r] = fminf(vmin[r], __shfl_xor(vmin[r], m, 32));
    }

    // Lanes 0 and 16 hold the chunk-min for rows r0+r and r0+r+8.
    if (l == 0) {
        const int base = r0 + (hi ? 8 : 0);
#pragma unroll
        for (int r = 0; r < 8; ++r) {
            // Distances are >= 0, so float bit pattern order == unsigned order.
            atomicMin(&rowminBits[base + r], __float_as_uint(vmin[r]));
        }
    }
}

// mean(min1) + mean(min2) -> scalar output.
__global__ void chamfer_finalize(const unsigned* __restrict__ r1, int n1,
                                 const unsigned* __restrict__ r2, int n2,
                                 float* __restrict__ out) {
    __shared__ float sdata[256];
    float s = 0.0f;
    const float inv1 = 1.0f / (float)n1;
    const float inv2 = 1.0f / (float)n2;
    for (int i = threadIdx.x; i < n1; i += 256) s += __uint_as_float(r1[i]) * inv1;
    for (int i = threadIdx.x; i < n2; i += 256) s += __uint_as_float(r2[i]) * inv2;
    sdata[threadIdx.x] = s;
    __syncthreads();
    for (int ofs = 128; ofs > 0; ofs >>= 1) {
        if ((int)threadIdx.x < ofs) sdata[threadIdx.x] += sdata[threadIdx.x + ofs];
        __syncthreads();
    }
    if (threadIdx.x == 0) out[0] = sdata[0];
}

extern "C" void kernel_launch(void* const* d_in, const int* in_sizes, int n_in,
                              void* d_out, int out_size, void* d_ws, size_t ws_size,
                              hipStream_t stream) {
    const float* p1 = (const float*)d_in[0];
    const float* p2 = (const float*)d_in[1];
    const int N = in_sizes[0] / 3;
    const int M = in_sizes[1] / 3;

    char* ws = (char*)d_ws;
    float2*   P1op = (float2*)ws;                                   // 16*N bytes
    float2*   P2op = (float2*)(ws + (size_t)N * 16);                // 16*M bytes
    unsigned* m1   = (unsigned*)(ws + (size_t)(N + M) * 16);        // 4*N bytes
    unsigned* m2   = (unsigned*)(ws + (size_t)(N + M) * 16 + (size_t)N * 4);
    float*    out  = (float*)d_out;

    const int nmax = N > M ? N : M;
    chamfer_prep<<<(nmax + 255) / 256, 256, 0, stream>>>(p1, N, p2, M, P1op, P2op, m1, m2);

    // Pass 1: for each point of cloud1, min dist to cloud2.
    const int units1 = (N / 16) * CCH;
    chamfer_rowmin<<<units1 / WPB, 256, 0, stream>>>(P1op, N, P2op, M, m1);

    // Pass 2: swapped roles -> for each point of cloud2, min dist to cloud1.
    const int units2 = (M / 16) * CCH;
    chamfer_rowmin<<<units2 / WPB, 256, 0, stream>>>(P2op, M, P1op, N, m2);

    chamfer_finalize<<<1, 256, 0, stream>>>(m1, N, m2, M, out);
}